// Net_28140625723913
// MI455X (gfx1250) — compile-verified
//
#include <hip/hip_runtime.h>
#include <math.h>

#define N_NODES 50000
#define N_EDGES 800000
#define F_IN    512
#define HIDDEN  128
#define NCLS    40
#define NCLS_PAD 48

// LDS pair-row strides (floats). Chosen so stride % 64 == 32 (dwords) ->
// half-wave 0 (pairRow p) and half-wave 1 (pairRow p+1) hit disjoint bank halves.
#define SW1_STRIDE 288   // 32 pair-rows * 288 = 9216 floats = 36 KB
#define SW2_STRIDE 96    // 64 pair-rows * 96  = 6144 floats = 24 KB (96 % 64 == 32)

typedef float v2f __attribute__((ext_vector_type(2)));
typedef float v8f __attribute__((ext_vector_type(8)));

static __device__ __forceinline__ v8f wmma_f32_k4(v2f a, v2f b, v8f c) {
    // V_WMMA_F32_16X16X4_F32 : D = A(16x4) * B(4x16) + C(16x16), fp32
    return __builtin_amdgcn_wmma_f32_16x16x4_f32(
        /*neg_a=*/false, a, /*neg_b=*/false, b,
        /*c_mod=*/(short)0, c, /*reuse_a=*/false, /*reuse_b=*/false);
}

// ---------------- degree / normalization ----------------

__global__ void gcn_init_deg(float* deg, int n) {
    int i = blockIdx.x * blockDim.x + threadIdx.x;
    if (i < n) deg[i] = 1.0f;               // self-loop contribution
}

__global__ void gcn_degree(const int* __restrict__ dst, float* __restrict__ deg, int nE) {
    int i = blockIdx.x * blockDim.x + threadIdx.x;
    if (i < nE) atomicAdd(deg + dst[i], 1.0f);
}

__global__ void gcn_rsqrt(float* deg, int n) {
    int i = blockIdx.x * blockDim.x + threadIdx.x;
    if (i < n) deg[i] = rsqrtf(deg[i]);     // deg >= 1 always
}

// ---------------- GEMM1: g1 = dinv .* (X @ W1); acc1 seeded with g1 ----------------
// Block = 256 threads = 8 waves; wave w computes rows [ (bid*8+w)*16 , +16 ) x all 128 cols.
// 50000 % 16 == 0 => every tile is fully valid or fully dead (wave-uniform).
__global__ __launch_bounds__(256) void gcn_gemm1_wmma(
    const float* __restrict__ X, const float* __restrict__ W,
    const float* __restrict__ dinv,
    float* __restrict__ g1, float* __restrict__ acc1) {

    __shared__ float sW[32 * SW1_STRIDE];   // K-pair-swizzled 64x128 chunk of W1

    const int tid  = threadIdx.x;
    const int wave = tid >> 5;
    const int lane = tid & 31;
    const int half = lane >> 4;             // K-pair selector for f32 A/B fragments
    const int mrow = lane & 15;             // row inside A-tile / col inside B,C tiles
    const int mbase = (blockIdx.x * 8 + wave) * 16;

    const int mg  = mbase + mrow;
    const int mgc = (mg < N_NODES) ? mg : 0;            // clamp; dead tiles never store
    const float* xrow = X + (size_t)mgc * F_IN;

    v8f acc[8] = {};

    for (int kc = 0; kc < F_IN; kc += 64) {
        // cooperative stage of W1[kc:kc+64, 0:128], swizzled into K-pairs:
        // dst = (r/2)*SW1_STRIDE + 2*c + (r&1)  -> B fragment = one aligned b64 load
        for (int idx = tid; idx < 64 * HIDDEN; idx += 256) {
            const int r = idx >> 7;
            const int c = idx & 127;
            sW[(r >> 1) * SW1_STRIDE + (c << 1) + (r & 1)] =
                W[(size_t)(kc + r) * HIDDEN + c];
        }
        __syncthreads();

        for (int k = 0; k < 64; k += 4) {
            const v2f a = *(const v2f*)(xrow + kc + k + 2 * half);   // (K, K+1)
            const float* bp = &sW[((k >> 1) + half) * SW1_STRIDE + (mrow << 1)];
#pragma unroll
            for (int nt = 0; nt < 8; ++nt) {
                const v2f b = *(const v2f*)(bp + (nt << 5));         // (K,n),(K+1,n)
                acc[nt] = wmma_f32_k4(a, b, acc[nt]);
            }
        }
        __syncthreads();
    }

    // epilogue: scale rows by dinv, write g1 and seed acc1 (self-loop term)
#pragma unroll
    for (int r = 0; r < 8; ++r) {
        const int M = mbase + r + 8 * half;
        if (M < N_NODES) {
            const float s = dinv[M];
#pragma unroll
            for (int nt = 0; nt < 8; ++nt) {
                const float v = s * acc[nt][r];
                const size_t o = (size_t)M * HIDDEN + nt * 16 + mrow;
                g1[o]   = v;
                acc1[o] = v;
            }
        }
    }
}

// ---------------- edge scatter, F = 128 (float4 per lane) ----------------
__global__ void gcn_scatter128(const float* __restrict__ g, float* __restrict__ acc,
                               const int* __restrict__ src, const int* __restrict__ dst,
                               int nE) {
    const long long idx   = (long long)blockIdx.x * blockDim.x + threadIdx.x;
    const long long total = (long long)nE * 32;
    if (idx >= total) return;
    const int e = (int)(idx >> 5);
    const int q = (int)(idx & 31);
    const int s = src[e];
    const int d = dst[e];
    const float4 v = *(const float4*)(g + (size_t)s * HIDDEN + q * 4);
    float* p = acc + (size_t)d * HIDDEN + q * 4;
    atomicAdd(p + 0, v.x);
    atomicAdd(p + 1, v.y);
    atomicAdd(p + 2, v.z);
    atomicAdd(p + 3, v.w);
}

// ---------------- finalize layer 1: h = relu(dinv .* acc1 + b1) ----------------
__global__ void gcn_finalize1(const float* __restrict__ acc1, const float* __restrict__ dinv,
                              const float* __restrict__ b1, float* __restrict__ hout) {
    const long long idx = (long long)blockIdx.x * blockDim.x + threadIdx.x;
    if (idx >= (long long)N_NODES * HIDDEN) return;
    const int n = (int)(idx >> 7);
    const int f = (int)(idx & 127);
    const float v = dinv[n] * acc1[idx] + b1[f];
    hout[idx] = fmaxf(v, 0.0f);
}

// ---------------- GEMM2: g2 = dinv .* (H @ W2); acc2 seeded with g2 ----------------
__global__ __launch_bounds__(256) void gcn_gemm2_wmma(
    const float* __restrict__ H, const float* __restrict__ W2,
    const float* __restrict__ dinv,
    float* __restrict__ g2, float* __restrict__ acc2) {

    __shared__ float sW[64 * SW2_STRIDE];   // K-pair-swizzled 128x48 (zero-padded) W2

    const int tid  = threadIdx.x;
    const int wave = tid >> 5;
    const int lane = tid & 31;
    const int half = lane >> 4;
    const int mrow = lane & 15;
    const int mbase = (blockIdx.x * 8 + wave) * 16;

    for (int idx = tid; idx < HIDDEN * NCLS_PAD; idx += 256) {
        const int r = idx / NCLS_PAD, c = idx % NCLS_PAD;
        sW[(r >> 1) * SW2_STRIDE + (c << 1) + (r & 1)] =
            (c < NCLS) ? W2[(size_t)r * NCLS + c] : 0.0f;
    }
    __syncthreads();

    const int mg  = mbase + mrow;
    const int mgc = (mg < N_NODES) ? mg : 0;
    const float* hrow = H + (size_t)mgc * HIDDEN;

    v8f acc[3] = {};

    for (int k = 0; k < HIDDEN; k += 4) {
        const v2f a = *(const v2f*)(hrow + k + 2 * half);
        const float* bp = &sW[((k >> 1) + half) * SW2_STRIDE + (mrow << 1)];
#pragma unroll
        for (int nt = 0; nt < 3; ++nt) {
            const v2f b = *(const v2f*)(bp + (nt << 5));
            acc[nt] = wmma_f32_k4(a, b, acc[nt]);
        }
    }

#pragma unroll
    for (int r = 0; r < 8; ++r) {
        const int M = mbase + r + 8 * half;
        if (M < N_NODES) {
            const float s = dinv[M];
#pragma unroll
            for (int nt = 0; nt < 3; ++nt) {
                const int c = nt * 16 + mrow;
                if (c < NCLS) {
                    const float v = s * acc[nt][r];
                    const size_t o = (size_t)M * NCLS + c;
                    g2[o]   = v;
                    acc2[o] = v;
                }
            }
        }
    }
}

// ---------------- edge scatter, F = 40 ----------------
__global__ void gcn_scatter40(const float* __restrict__ g, float* __restrict__ acc,
                              const int* __restrict__ src, const int* __restrict__ dst,
                              int nE) {
    const long long idx   = (long long)blockIdx.x * blockDim.x + threadIdx.x;
    const long long total = (long long)nE * NCLS;
    if (idx >= total) return;
    const int e = (int)(idx / NCLS);
    const int c = (int)(idx % NCLS);
    atomicAdd(acc + (size_t)dst[e] * NCLS + c, g[(size_t)src[e] * NCLS + c]);
}

// ---------------- finalize layer 2: bias + log_softmax ----------------
__global__ void gcn_finalize2(const float* __restrict__ acc2, const float* __restrict__ dinv,
                              const float* __restrict__ b2, float* __restrict__ out, int n) {
    const int i = blockIdx.x * blockDim.x + threadIdx.x;
    if (i >= n) return;
    float v[NCLS];
    const float s = dinv[i];
    float mx = -INFINITY;
#pragma unroll
    for (int c = 0; c < NCLS; ++c) {
        v[c] = s * acc2[(size_t)i * NCLS + c] + b2[c];
        mx = fmaxf(mx, v[c]);
    }
    float sum = 0.0f;
#pragma unroll
    for (int c = 0; c < NCLS; ++c) sum += expf(v[c] - mx);
    const float lse = logf(sum) + mx;
#pragma unroll
    for (int c = 0; c < NCLS; ++c) out[(size_t)i * NCLS + c] = v[c] - lse;
}

// ---------------- host-side orchestration ----------------

extern "C" void kernel_launch(void* const* d_in, const int* in_sizes, int n_in,
                              void* d_out, int out_size, void* d_ws, size_t ws_size,
                              hipStream_t stream) {
    const float* x  = (const float*)d_in[0];
    const int*   ei = (const int*)  d_in[1];   // [2, E]: row 0 = src, row 1 = dst
    const float* W1 = (const float*)d_in[2];
    const float* b1 = (const float*)d_in[3];
    const float* W2 = (const float*)d_in[4];
    const float* b2 = (const float*)d_in[5];
    float* out = (float*)d_out;

    const int* src = ei;
    const int* dst = ei + N_EDGES;

    // workspace layout
    float* dinv = (float*)d_ws;                                   // N floats
    float* bufA = (float*)((char*)d_ws + (1 << 18));              // N*128 : g1, then hidden activations
    float* bufB = bufA + (size_t)N_NODES * HIDDEN;                // N*128 : acc1, then reused for g2/acc2
    float* g2   = bufB;                                           // N*40
    float* acc2 = bufB + (size_t)N_NODES * NCLS;                  // N*40

    const int T = 256;

    // normalization: deg -> dinv
    gcn_init_deg<<<(N_NODES + T - 1) / T, T, 0, stream>>>(dinv, N_NODES);
    gcn_degree  <<<(N_EDGES + T - 1) / T, T, 0, stream>>>(dst, dinv, N_EDGES);
    gcn_rsqrt   <<<(N_NODES + T - 1) / T, T, 0, stream>>>(dinv, N_NODES);

    // layer 1
    const int gemmBlocks = (N_NODES + 127) / 128;                 // 8 waves * 16 rows per block
    gcn_gemm1_wmma<<<gemmBlocks, T, 0, stream>>>(x, W1, dinv, bufA, bufB);

    {
        const long long total = (long long)N_EDGES * 32;
        gcn_scatter128<<<(int)((total + T - 1) / T), T, 0, stream>>>(bufA, bufB, src, dst, N_EDGES);
    }
    {
        const long long total = (long long)N_NODES * HIDDEN;
        gcn_finalize1<<<(int)((total + T - 1) / T), T, 0, stream>>>(bufB, dinv, b1, bufA);
    }

    // layer 2
    gcn_gemm2_wmma<<<gemmBlocks, T, 0, stream>>>(bufA, W2, dinv, g2, acc2);
    {
        const long long total = (long long)N_EDGES * NCLS;
        gcn_scatter40<<<(int)((total + T - 1) / T), T, 0, stream>>>(g2, acc2, src, dst, N_EDGES);
    }
    gcn_finalize2<<<(N_NODES + T - 1) / T, T, 0, stream>>>(acc2, dinv, b2, out, N_NODES);
}